// CreateTangentImages_58463094833211
// MI455X (gfx1250) — compile-verified
//
#include <hip/hip_runtime.h>

// Bispherical (bilinear) resampling of an equirect image onto tangent grids.
// x:          [B=2, C=3, H=2048, W=4096]  float32
// sample_map: [N=80, 256, 256, 2]         float32 (x,y) equirect coords
// out:        [B, C, N, 256, 256]         float32
//
// Memory-bound gather kernel. Strategy:
//  - 2 points per thread: one coalesced 16B NT load of coords,
//    one 8B NT store per plane (aligned since P is even).
//  - Streams (coords in, output out) use nontemporal hints so the 192 MB L2
//    is preserved for the 201 MB gather image (regular-temporal gathers).
//  - Gather offsets computed once per point, reused across all 6 planes
//    (scalar plane base + vector offset addressing).

typedef float v4f __attribute__((ext_vector_type(4)));
typedef float v2f __attribute__((ext_vector_type(2)));

namespace {
constexpr int  kH      = 2048;
constexpr int  kW      = 4096;               // power of two -> wrap via AND
constexpr int  kFaces  = 80;
constexpr int  kGrid   = 256;
constexpr int  kB      = 2;
constexpr int  kC      = 3;
constexpr int  kPlanes = kB * kC;                          // 6
constexpr long kPlaneElems = (long)kH * kW;                // 8,388,608
constexpr long kP      = (long)kFaces * kGrid * kGrid;     // 5,242,880
constexpr int  kBlock  = 256;                              // 8 waves (wave32)
}  // namespace

struct Pt {
  int   o00, o01, o10, o11;   // element offsets into one [H,W] plane
  float w00, w01, w10, w11;   // bilinear weights
};

__device__ __forceinline__ Pt prep_point(float sx, float sy) {
  Pt p;
  const float xf = floorf(sx);
  const float yf = floorf(sy);
  const float wx = sx - xf;
  const float wy = sy - yf;

  int x0 = ((int)xf) & (kW - 1);          // longitude wraps (spherical seam)
  int x1 = (x0 + 1) & (kW - 1);
  int y0 = (int)yf;
  y0 = (y0 < 0) ? 0 : ((y0 > kH - 1) ? (kH - 1) : y0);
  int y1 = (y0 + 1 > kH - 1) ? (kH - 1) : (y0 + 1);  // latitude clamps (poles)

  const int r0 = y0 * kW;
  const int r1 = y1 * kW;
  p.o00 = r0 + x0;
  p.o01 = r0 + x1;
  p.o10 = r1 + x0;
  p.o11 = r1 + x1;

  const float iwx = 1.0f - wx;
  const float iwy = 1.0f - wy;
  p.w00 = iwx * iwy;
  p.w01 = wx * iwy;
  p.w10 = iwx * wy;
  p.w11 = wx * wy;
  return p;
}

__global__ __launch_bounds__(kBlock) void bisph_resample(
    const float* __restrict__ x, const float* __restrict__ smap,
    float* __restrict__ out) {
  const long gid = (long)blockIdx.x * kBlock + threadIdx.x;  // 2 points/thread
  const long q = gid * 2;
  if (q >= kP) return;

  // Both coordinate pairs in one coalesced 16B nontemporal load (streamed once).
  const v4f c = __builtin_nontemporal_load(((const v4f*)smap) + gid);

  const Pt a = prep_point(c.x, c.y);
  const Pt b = prep_point(c.z, c.w);

  // 6 planes x 8 gathers, fully unrolled: ~48 loads in flight per thread.
  // Plane base is scalar-uniform; per-lane offsets are reused across planes.
#pragma unroll
  for (int pl = 0; pl < kPlanes; ++pl) {
    const float* __restrict__ plane = x + (long)pl * kPlaneElems;

    const float va = plane[a.o00] * a.w00 + plane[a.o01] * a.w01 +
                     plane[a.o10] * a.w10 + plane[a.o11] * a.w11;
    const float vb = plane[b.o00] * b.w00 + plane[b.o01] * b.w01 +
                     plane[b.o10] * b.w10 + plane[b.o11] * b.w11;

    // 8-byte-aligned nontemporal b64 store: output streamed once, keep it
    // out of L2 so the gather image stays resident.
    v2f v;
    v.x = va;
    v.y = vb;
    __builtin_nontemporal_store(v, ((v2f*)(out + (long)pl * kP)) + gid);
  }
}

extern "C" void kernel_launch(void* const* d_in, const int* in_sizes, int n_in,
                              void* d_out, int out_size, void* d_ws,
                              size_t ws_size, hipStream_t stream) {
  const float* x    = (const float*)d_in[0];   // [2,3,2048,4096]
  const float* smap = (const float*)d_in[1];   // [80,256,256,2]
  float* out        = (float*)d_out;           // [2,3,80,256,256]

  const long nThreads = kP / 2;                          // 2,621,440
  const int  nBlocks  = (int)((nThreads + kBlock - 1) / kBlock);  // 10240
  bisph_resample<<<nBlocks, kBlock, 0, stream>>>(x, smap, out);
}